// ExpertChoiceMoEGate_64003602645070
// MI455X (gfx1250) — compile-verified
//
#include <hip/hip_runtime.h>

#define NB 2
#define NS 4096
#define ND 2048
#define NE 16
#define NC 512

typedef __attribute__((ext_vector_type(2))) float v2f;
typedef __attribute__((ext_vector_type(4))) float v4f;
typedef __attribute__((ext_vector_type(8))) float v8f;

// ---------------------------------------------------------------------------
// DPP16 butterfly reductions over 16-lane groups (no LDS pipe usage).
// quad_perm[1,0,3,2]=0xB1 (xor1), quad_perm[2,3,0,1]=0x4E (xor2),
// row_half_mirror=0x141 (combines quads within 8), row_mirror=0x140
// (combines 8-groups within 16). All lanes active in this kernel.
// ---------------------------------------------------------------------------
template <int CTRL>
__device__ __forceinline__ float dpp_perm(float v) {
  return __int_as_float(__builtin_amdgcn_update_dpp(
      0, __float_as_int(v), CTRL, 0xF, 0xF, true));
}

__device__ __forceinline__ float redmax16(float v) {
  v = fmaxf(v, dpp_perm<0xB1>(v));
  v = fmaxf(v, dpp_perm<0x4E>(v));
  v = fmaxf(v, dpp_perm<0x141>(v));
  v = fmaxf(v, dpp_perm<0x140>(v));
  return v;
}

__device__ __forceinline__ float redsum16(float v) {
  v += dpp_perm<0xB1>(v);
  v += dpp_perm<0x4E>(v);
  v += dpp_perm<0x141>(v);
  v += dpp_perm<0x140>(v);
  return v;
}

// ---------------------------------------------------------------------------
// Kernel 1: logits = x @ router via V_WMMA_F32_16X16X4_F32, then softmax over
// the 16 experts (DPP reduction within each 16-lane half-wave), store
// affinity transposed as (B, E, S) into workspace.
//
// One wave32 computes one 16-token x 16-expert tile. 128 threads = 4 waves.
// A-layout (16x4 f32): lanes 0-15 -> M=0..15 {K0,K1}; lanes 16-31 -> {K2,K3}.
// B-layout (4x16 f32): lanes 0-15 -> N=0..15 {K0,K1}; lanes 16-31 -> {K2,K3}.
// C/D (16x16 f32, 8 VGPRs): lane<16 -> N=lane, M=r; lane>=16 -> N=lane-16, M=r+8.
// ---------------------------------------------------------------------------
__global__ __launch_bounds__(128) void moe_logits_softmax(
    const float* __restrict__ x,       // (B,S,D) row-major
    const float* __restrict__ router,  // (D,E) row-major
    float* __restrict__ aff)           // (B,E,S)
{
  const int wave = threadIdx.x >> 5;
  const int lane = threadIdx.x & 31;
  const int tile = blockIdx.x * 4 + wave;   // 512 tiles of 16 tokens
  const int m0   = tile * 16;               // flat token base (b*S + s)
  const int half = lane >> 4;               // 0 or 1
  const int mr   = lane & 15;

  const float* __restrict__ xrow = x + (size_t)(m0 + mr) * ND + 2 * half;
  const float* __restrict__ brow = router + (size_t)(2 * half) * NE + mr;

  v8f acc = {};
#pragma unroll 8
  for (int k = 0; k < ND; k += 4) {
    v2f a, b;
    a.x = xrow[k + 0];                 // x[m][k + 2*half]
    a.y = xrow[k + 1];                 // x[m][k + 2*half + 1]
    b.x = brow[(size_t)k * NE];        // router[(k + 2*half)][n]
    b.y = brow[(size_t)(k + 1) * NE];  // router[(k + 2*half + 1)][n]
    acc = __builtin_amdgcn_wmma_f32_16x16x4_f32(
        /*neg_a=*/false, a, /*neg_b=*/false, b,
        /*c_mod=*/(short)0, acc, /*reuse_a=*/false, /*reuse_b=*/false);
  }

  // acc[r] = logits[token m0 + r + 8*half][expert mr]; softmax over experts
  // = reduction across the 16 lanes of this half-wave (pure VALU via DPP).
  float p[8];
#pragma unroll
  for (int r = 0; r < 8; ++r) {
    float v  = acc[r];
    float mx = redmax16(v);
    float e  = expf(v - mx);
    float sm = redsum16(e);
    p[r] = e / sm;
  }

  // Store transposed: aff[b][e = mr][s = s0 + r + 8*half]
  const int b  = m0 / NS;
  const int s0 = m0 % NS;
  float* out = aff + ((size_t)b * NE + mr) * NS + s0 + 8 * half;
  v4f lo = {p[0], p[1], p[2], p[3]};
  v4f hi = {p[4], p[5], p[6], p[7]};
  *(v4f*)(out + 0) = lo;   // 32B aligned: s0 + 8*half is a multiple of 8
  *(v4f*)(out + 4) = hi;
}

// ---------------------------------------------------------------------------
// Kernel 2: per (b,e), sorted top-512 of 4096 affinities via in-LDS bitonic
// sort with (value desc, index asc) comparator -- matches jax.lax.top_k
// ordering and tie-break. One workgroup (256 threads) per (b,e); 32KB LDS.
// Then writes gating, index, and the block's full 8MB one-hot dispatch slab
// in a single pass with non-temporal b128 stores (256MB total > L2; stream).
// ---------------------------------------------------------------------------
__global__ __launch_bounds__(256) void topk_bitonic(
    const float* __restrict__ aff,   // (B,E,S)
    float* __restrict__ gating,      // (B,E,C)
    float* __restrict__ dispatch,    // (B,E,C,S)
    int*   __restrict__ index)       // (B,E,C)
{
  __shared__ float sv[NS];
  __shared__ int   si[NS];
  const int be = blockIdx.x;         // 0..31
  const float* __restrict__ src = aff + (size_t)be * NS;

  for (int i = threadIdx.x; i < NS; i += 256) { sv[i] = src[i]; si[i] = i; }
  __syncthreads();

  for (int k = 2; k <= NS; k <<= 1) {
    for (int j = k >> 1; j > 0; j >>= 1) {
      for (int t = threadIdx.x; t < NS / 2; t += 256) {
        int i  = 2 * t - (t & (j - 1));
        int pp = i + j;
        float vi = sv[i], vp = sv[pp];
        int   ii = si[i], ip = si[pp];
        // "i should precede pp" in descending order (lower index wins ties)
        bool keep = (vi > vp) || (vi == vp && ii < ip);
        bool dir  = ((i & k) == 0);
        if (dir != keep) {
          sv[i] = vp; sv[pp] = vi;
          si[i] = ip; si[pp] = ii;
        }
      }
      __syncthreads();
    }
  }

  // gating + index for the top C
  float* g  = gating + (size_t)be * NC;
  int*   ix = index  + (size_t)be * NC;
  for (int c = threadIdx.x; c < NC; c += 256) {
    g[c]  = sv[c];
    ix[c] = si[c];
  }

  // One-hot dispatch slab: row c is all zeros except a 1.0 at token si[c].
  // 512 rows x 4096 floats = 8MB per block, written once, non-temporally.
  v4f* dp = (v4f*)(dispatch + (size_t)be * NC * NS);   // 1024 v4f per row
  for (int c = 0; c < NC; ++c) {
    const int t   = si[c];           // LDS broadcast
    const int tq  = t >> 2;          // which v4f holds the one
    const int tl  = t & 3;
    v4f* row = dp + (size_t)c * (NS / 4);
    for (int j = threadIdx.x; j < NS / 4; j += 256) {
      v4f z = {0.f, 0.f, 0.f, 0.f};
      if (j == tq) z[tl] = 1.0f;
      __builtin_nontemporal_store(z, row + j);
    }
  }
}

// ---------------------------------------------------------------------------
extern "C" void kernel_launch(void* const* d_in, const int* in_sizes, int n_in,
                              void* d_out, int out_size, void* d_ws, size_t ws_size,
                              hipStream_t stream)
{
  const float* x      = (const float*)d_in[0];   // (B,S,D)
  const float* router = (const float*)d_in[1];   // (D,E)

  float* aff = (float*)d_ws;                          // B*E*S = 512KB scratch

  float* gating   = (float*)d_out;                    // 2*16*512      floats
  float* dispatch = gating + (size_t)NB * NE * NC;    // 2*16*512*4096 floats
  int*   index    = (int*)(dispatch + (size_t)NB * NE * NC * NS); // int32 bits

  // 1) logits + softmax -> affinity_t (B,E,S)
  const int tiles = (NB * NS) / 16;                   // 512 tiles
  moe_logits_softmax<<<tiles / 4, 128, 0, stream>>>(x, router, aff);

  // 2) sorted top-C per (b,e): gating, index, one-hot dispatch (fused write)
  topk_bitonic<<<NB * NE, 256, 0, stream>>>(aff, gating, dispatch, index);
}